// GraphConv_72799695667427
// MI455X (gfx1250) — compile-verified
//
#include <hip/hip_runtime.h>
#include <hip/hip_bf16.h>

#define N_USERS 50000
#define N_ENT   100000
#define N_INT   5
#define EMB     128
#define N_REL   20
#define N_EDGES 1000000
#define NNZ     1000000

typedef __attribute__((ext_vector_type(2))) float v2f;
typedef __attribute__((ext_vector_type(8))) float v8f;

// ---------------- init: res_e/res_u = inputs, cur = inputs, zero cor ----------------
__global__ void k_init(const float* __restrict__ Ein, const float* __restrict__ Uin,
                       float* __restrict__ Re, float* __restrict__ Ru,
                       float* __restrict__ Ecur, float* __restrict__ Ucur,
                       float* __restrict__ cor) {
  size_t idx = (size_t)blockIdx.x * 256 + threadIdx.x;   // grid covers N_ENT*EMB exactly
  float e = Ein[idx];
  Re[idx] = e;
  Ecur[idx] = e;
  if (idx < (size_t)N_USERS * EMB) {
    float u = Uin[idx];
    Ru[idx] = u;
    Ucur[idx] = u;
  }
  if (idx == 0) *cor = 0.0f;
}

// ---------------- intent2 = (intent_agg + intent_emb)/2, hop-invariant ----------------
__global__ void k_intent2(const float* __restrict__ IE, const float* __restrict__ R,
                          float* __restrict__ I2) {
  __shared__ float dots[40];
  __shared__ float att[5][20];
  int t = threadIdx.x;  // 128 threads
  // 40 (intent, relation) dot products
  if (t < 40) {
    int k, r;
    if (t < 20)      { k = 0; r = t; }
    else if (t < 24) { k = 1; r = t - 20; }
    else if (t < 28) { k = 2; r = t - 24; }
    else if (t < 32) { k = 3; r = t - 28; }
    else             { k = 4; r = t - 32; }
    int start = (k == 0) ? 0 : (k == 4) ? 12 : (k - 1) * 4;
    int row = start + r;
    float s = 0.0f;
    for (int d = 0; d < EMB; d++) s += IE[k * EMB + d] * R[row * EMB + d];
    dots[t] = s;
  }
  __syncthreads();
  if (t < 5) {
    int off = (t == 0) ? 0 : (t == 1) ? 20 : (t == 2) ? 24 : (t == 3) ? 28 : 32;
    int n   = (t == 0) ? 20 : (t == 4) ? 8 : 4;
    float m = -3.4e38f;
    for (int r = 0; r < n; r++) m = fmaxf(m, dots[off + r]);
    float s = 0.0f;
    for (int r = 0; r < n; r++) { float e = expf(dots[off + r] - m); att[t][r] = e; s += e; }
    for (int r = 0; r < n; r++) att[t][r] /= s;
  }
  __syncthreads();
  // t = embedding dim
  for (int k = 0; k < 5; k++) {
    int start = (k == 0) ? 0 : (k == 4) ? 12 : (k - 1) * 4;
    int n     = (k == 0) ? 20 : (k == 4) ? 8 : 4;
    float agg = 0.0f;
    for (int r = 0; r < n; r++) agg += att[k][r] * R[(start + r) * EMB + t];
    agg /= (float)n;
    I2[k * EMB + t] = 0.5f * (agg + IE[k * EMB + t]);
  }
}

// ---------------- distance correlation (one pair per block, 10 blocks) ----------------
__global__ void k_cor(const float* __restrict__ IE, float* __restrict__ cor_out) {
  __shared__ float t1[128], t2[128], am[128], bm[128];
  __shared__ float red[256];
  __shared__ float amean_s, bmean_s, rAB, rAA;
  const int pi[10] = {0,0,0,0,1,1,1,2,2,3};
  const int pj[10] = {1,2,3,4,2,3,4,3,4,4};
  int i = pi[blockIdx.x], j = pj[blockIdx.x];
  int t = threadIdx.x;
  if (t < 128) { t1[t] = IE[i * EMB + t]; t2[t] = IE[j * EMB + t]; }
  __syncthreads();
  if (t < 128) {
    float sa = 0.0f, sb = 0.0f;
    for (int q = 0; q < 128; q++) {
      float da = t1[t] - t1[q]; sa += sqrtf(da * da + 1e-8f);
      float db = t2[t] - t2[q]; sb += sqrtf(db * db + 1e-8f);
    }
    am[t] = sa / 128.0f; bm[t] = sb / 128.0f;
  }
  __syncthreads();
  if (t == 0) {
    float sa = 0.0f, sb = 0.0f;
    for (int q = 0; q < 128; q++) { sa += am[q]; sb += bm[q]; }
    amean_s = sa / 128.0f; bmean_s = sb / 128.0f;
  }
  __syncthreads();
  float ab = 0.0f, aa = 0.0f, bb = 0.0f;
  for (int idx = t; idx < 16384; idx += 256) {
    int r = idx >> 7, c = idx & 127;
    float da = t1[r] - t1[c];
    float a = sqrtf(da * da + 1e-8f) - am[r] - am[c] + amean_s;
    float db = t2[r] - t2[c];
    float b = sqrtf(db * db + 1e-8f) - bm[r] - bm[c] + bmean_s;
    ab += a * b; aa += a * a; bb += b * b;
  }
  // three block reductions
  red[t] = ab; __syncthreads();
  for (int s = 128; s > 0; s >>= 1) { if (t < s) red[t] += red[t + s]; __syncthreads(); }
  if (t == 0) rAB = red[0];
  __syncthreads();
  red[t] = aa; __syncthreads();
  for (int s = 128; s > 0; s >>= 1) { if (t < s) red[t] += red[t + s]; __syncthreads(); }
  if (t == 0) rAA = red[0];
  __syncthreads();
  red[t] = bb; __syncthreads();
  for (int s = 128; s > 0; s >>= 1) { if (t < s) red[t] += red[t + s]; __syncthreads(); }
  if (t == 0) {
    const float d2 = 16384.0f;
    float dAB = sqrtf(fmaxf(rAB / d2, 0.0f) + 1e-8f);
    float dAA = sqrtf(fmaxf(rAA / d2, 0.0f) + 1e-8f);
    float dBB = sqrtf(fmaxf(red[0] / d2, 0.0f) + 1e-8f);
    atomicAdd(cor_out, dAB / sqrtf(dAA * dBB + 1e-8f));
  }
}

// ---------------- entity scatter: msg = e[tail]*r_emb[type-1] -> e_agg[head] ----------------
__global__ void __launch_bounds__(256)
k_scatter_e(const int* __restrict__ EI, const int* __restrict__ ET,
            const float* __restrict__ Ecur, const float* __restrict__ R,
            float* __restrict__ Eagg, float* __restrict__ DEG, int nEdges) {
  int lane = threadIdx.x & 31, wave = threadIdx.x >> 5;
  int edge = blockIdx.x * 8 + wave;
  if (edge >= nEdges) return;
  int head = EI[edge];
  int tail = EI[N_EDGES + edge];
  int ty = ET[edge] - 1;
  int d = lane * 4;
  float4 ev = *(const float4*)(Ecur + (size_t)tail * EMB + d);
  float4 rv = *(const float4*)(R + (size_t)ty * EMB + d);
  float* dst = Eagg + (size_t)head * EMB + d;
  atomicAdd(dst + 0, ev.x * rv.x);
  atomicAdd(dst + 1, ev.y * rv.y);
  atomicAdd(dst + 2, ev.z * rv.z);
  atomicAdd(dst + 3, ev.w * rv.w);
  if (lane == 0) atomicAdd(DEG + head, 1.0f);
}

// ---------------- user scatter: vals[i]*e[col] -> u_agg[row] ----------------
__global__ void __launch_bounds__(256)
k_scatter_u(const int* __restrict__ IR, const int* __restrict__ IC,
            const float* __restrict__ IV, const float* __restrict__ Ecur,
            float* __restrict__ Uagg, int nnz) {
  int lane = threadIdx.x & 31, wave = threadIdx.x >> 5;
  int i = blockIdx.x * 8 + wave;
  if (i >= nnz) return;
  int row = IR[i];
  int col = IC[i];
  float val = IV[i];
  int d = lane * 4;
  float4 ev = *(const float4*)(Ecur + (size_t)col * EMB + d);
  float* dst = Uagg + (size_t)row * EMB + d;
  atomicAdd(dst + 0, val * ev.x);
  atomicAdd(dst + 1, val * ev.y);
  atomicAdd(dst + 2, val * ev.z);
  atomicAdd(dst + 3, val * ev.w);
}

// ---------------- score = softmax(u @ intent2.T) via V_WMMA_F32_16X16X4_F32 ----------------
// Wave handles a 16-user tile. A: 16x4 f32 (lanes 0-15 K={0,1}, 16-31 K={2,3});
// B: 4x16 f32 (VGPR0 rows K=0/K=2, VGPR1 rows K=1/K=3, N striped across half-lanes);
// C/D: 16x16 f32, VGPR v holds M=v (lanes 0-15) / M=v+8 (lanes 16-31).
// B panel (intent2^T padded N=5->16 with zero columns) is staged in LDS so the
// fragment reads are unconditional ds_load_b64 (no divergent exec-masked loads).
__global__ void __launch_bounds__(256)
k_score(const float* __restrict__ U, const float* __restrict__ I2,
        float* __restrict__ S, int nU) {
  __shared__ float sB[16 * EMB];     // 8 KB: rows 0-4 = intent2, rows 5-15 = 0
  __shared__ float slog[8][16][16];  // 8 KB logit staging
  // cooperative fill of the padded B panel (I2 is [5][128] row-major)
  for (int idx = threadIdx.x; idx < 16 * EMB; idx += 256)
    sB[idx] = (idx < N_INT * EMB) ? I2[idx] : 0.0f;
  __syncthreads();

  int lane = threadIdx.x & 31, wave = threadIdx.x >> 5;
  int base = (blockIdx.x * 8 + wave) * 16;
  int halb = lane >> 4;        // 0: lanes 0-15, 1: lanes 16-31
  int l15  = lane & 15;
  int rowA = base + l15;
  int rc = rowA < nU ? rowA : nU - 1;   // keep EXEC all-1s; clamp reads
  int kof = halb * 2;
  const float* urow = U + (size_t)rc * EMB + kof;
  const float* brow = sB + (size_t)l15 * EMB + kof;
  v8f acc = {0.f, 0.f, 0.f, 0.f, 0.f, 0.f, 0.f, 0.f};
  #pragma unroll 4
  for (int k0 = 0; k0 < EMB; k0 += 4) {
    v2f a, b;
    a.x = urow[k0];
    a.y = urow[k0 + 1];
    b.x = brow[k0];
    b.y = brow[k0 + 1];
    acc = __builtin_amdgcn_wmma_f32_16x16x4_f32(false, a, false, b, (short)0, acc,
                                                false, false);
  }
  #pragma unroll
  for (int v = 0; v < 8; v++)
    slog[wave][v + halb * 8][l15] = acc[v];
  __syncthreads();
  if (lane < 16) {
    int user = base + lane;
    if (user < nU) {
      float l0 = slog[wave][lane][0], l1 = slog[wave][lane][1], l2 = slog[wave][lane][2];
      float l3 = slog[wave][lane][3], l4 = slog[wave][lane][4];
      float m = fmaxf(fmaxf(fmaxf(l0, l1), fmaxf(l2, l3)), l4);
      float e0 = expf(l0 - m), e1 = expf(l1 - m), e2 = expf(l2 - m);
      float e3 = expf(l3 - m), e4 = expf(l4 - m);
      float inv = 1.0f / (e0 + e1 + e2 + e3 + e4);
      float* sp = S + (size_t)user * 5;
      sp[0] = e0 * inv; sp[1] = e1 * inv; sp[2] = e2 * inv; sp[3] = e3 * inv; sp[4] = e4 * inv;
    }
  }
}

// ---------------- user update: u = l2norm(u_agg*(1+score@intent2)); res_u += u ----------------
__global__ void __launch_bounds__(256)
k_update_u(const float* __restrict__ UA, const float* __restrict__ S,
           const float* __restrict__ I2, float* __restrict__ Ucur,
           float* __restrict__ Ru, int nU) {
  int lane = threadIdx.x & 31, wave = threadIdx.x >> 5;
  int user = blockIdx.x * 8 + wave;
  if (user >= nU) return;
  const float* sp = S + (size_t)user * 5;
  float s0 = sp[0], s1 = sp[1], s2 = sp[2], s3 = sp[3], s4 = sp[4];
  int d = lane * 4;
  size_t o = (size_t)user * EMB + d;
  float v[4];
  #pragma unroll
  for (int j = 0; j < 4; j++) {
    int dd = d + j;
    float boost = s0 * I2[dd] + s1 * I2[EMB + dd] + s2 * I2[2 * EMB + dd] +
                  s3 * I2[3 * EMB + dd] + s4 * I2[4 * EMB + dd];
    v[j] = UA[o + j] * (1.0f + boost);
  }
  float ss = v[0] * v[0] + v[1] * v[1] + v[2] * v[2] + v[3] * v[3];
  #pragma unroll
  for (int off = 16; off > 0; off >>= 1) ss += __shfl_xor(ss, off, 32);
  float inv = 1.0f / fmaxf(sqrtf(ss), 1e-12f);
  #pragma unroll
  for (int j = 0; j < 4; j++) {
    float nv = v[j] * inv;
    Ucur[o + j] = nv;
    Ru[o + j] += nv;
  }
}

// ---------------- entity update: e = l2norm(e_agg/deg); res_e += e ----------------
__global__ void __launch_bounds__(256)
k_update_e(const float* __restrict__ EA, const float* __restrict__ DEG,
           float* __restrict__ Ecur, float* __restrict__ Re, int nE) {
  int lane = threadIdx.x & 31, wave = threadIdx.x >> 5;
  int ent = blockIdx.x * 8 + wave;
  if (ent >= nE) return;
  float dg = DEG[ent];
  float invd = dg > 0.0f ? 1.0f / dg : 0.0f;
  int d = lane * 4;
  size_t o = (size_t)ent * EMB + d;
  float v[4];
  #pragma unroll
  for (int j = 0; j < 4; j++) v[j] = EA[o + j] * invd;
  float ss = v[0] * v[0] + v[1] * v[1] + v[2] * v[2] + v[3] * v[3];
  #pragma unroll
  for (int off = 16; off > 0; off >>= 1) ss += __shfl_xor(ss, off, 32);
  float inv = 1.0f / fmaxf(sqrtf(ss), 1e-12f);
  #pragma unroll
  for (int j = 0; j < 4; j++) {
    float nv = v[j] * inv;
    Ecur[o + j] = nv;
    Re[o + j] += nv;
  }
}

extern "C" void kernel_launch(void* const* d_in, const int* in_sizes, int n_in,
                              void* d_out, int out_size, void* d_ws, size_t ws_size,
                              hipStream_t stream) {
  const float* user_emb   = (const float*)d_in[0];
  const float* entity_emb = (const float*)d_in[1];
  const float* intent_emb = (const float*)d_in[2];
  const int*   edge_index = (const int*)d_in[3];
  const int*   edge_type  = (const int*)d_in[4];
  const int*   irows      = (const int*)d_in[5];
  const int*   icols      = (const int*)d_in[6];
  const float* ivals      = (const float*)d_in[7];
  const float* r_emb      = (const float*)d_in[8];

  float* out   = (float*)d_out;
  float* res_e = out;                                  // N_ENT*EMB
  float* res_u = out + (size_t)N_ENT * EMB;            // N_USERS*EMB
  float* cor   = out + (size_t)N_ENT * EMB + (size_t)N_USERS * EMB;  // scalar

  float* ws    = (float*)d_ws;
  float* e_cur = ws;                                             // 12.8M
  float* u_cur = e_cur + (size_t)N_ENT * EMB;                    // 6.4M
  float* e_agg = u_cur + (size_t)N_USERS * EMB;                  // 12.8M
  float* u_agg = e_agg + (size_t)N_ENT * EMB;                    // 6.4M
  float* deg   = u_agg + (size_t)N_USERS * EMB;                  // 100K
  float* int2  = deg + (size_t)N_ENT;                            // 640
  float* score = int2 + (size_t)N_INT * EMB;                     // 250K

  // residual init + cor slot zero
  k_init<<<(N_ENT * EMB) / 256, 256, 0, stream>>>(entity_emb, user_emb, res_e, res_u,
                                                  e_cur, u_cur, cor);
  // hop-invariant intent2 and the correlation regularizer
  k_intent2<<<1, 128, 0, stream>>>(intent_emb, r_emb, int2);
  k_cor<<<10, 256, 0, stream>>>(intent_emb, cor);

  for (int hop = 0; hop < 2; hop++) {
    hipMemsetAsync(e_agg, 0, (size_t)N_ENT * EMB * sizeof(float), stream);
    hipMemsetAsync(u_agg, 0, (size_t)N_USERS * EMB * sizeof(float), stream);
    hipMemsetAsync(deg, 0, (size_t)N_ENT * sizeof(float), stream);

    k_scatter_e<<<(N_EDGES + 7) / 8, 256, 0, stream>>>(edge_index, edge_type, e_cur,
                                                       r_emb, e_agg, deg, N_EDGES);
    k_scatter_u<<<(NNZ + 7) / 8, 256, 0, stream>>>(irows, icols, ivals, e_cur, u_agg, NNZ);
    k_score<<<(N_USERS + 127) / 128, 256, 0, stream>>>(u_cur, int2, score, N_USERS);
    k_update_u<<<(N_USERS + 7) / 8, 256, 0, stream>>>(u_agg, score, int2, u_cur, res_u,
                                                      N_USERS);
    k_update_e<<<(N_ENT + 7) / 8, 256, 0, stream>>>(e_agg, deg, e_cur, res_e, N_ENT);
  }
}